// RegProxyAffinityHead_34265249087972
// MI455X (gfx1250) — compile-verified
//
#include <hip/hip_runtime.h>
#include <hip/hip_bf16.h>
#include <math.h>

#define H_   24
#define W_   24
#define N_   576
#define D_   1024
#define B_   32
#define KSEL 256
#define MTOK 128

typedef __attribute__((ext_vector_type(2))) float v2f;
typedef __attribute__((ext_vector_type(8))) float v8f;

// ---------------------------------------------------------------------------
// Stage 1: depthwise 3x3 conv (SAME, zero pad) + pointwise (D->9) + softmax
// + immediate argmax over clipped-neighbor columns -> parent pointer.
// One block per (b, pixel); 256 threads, 4 channels/thread.
// ---------------------------------------------------------------------------
__global__ __launch_bounds__(256)
void affinity_parent_kernel(const float* __restrict__ tok2d,
                            const float* __restrict__ dw_w,
                            const float* __restrict__ pw_w,
                            const float* __restrict__ pw_b,
                            int* __restrict__ parent)
{
    const int bn = blockIdx.x;
    const int b  = bn / N_;
    const int n  = bn % N_;
    const int y  = n / W_;
    const int x  = n % W_;
    const int t  = threadIdx.x;
    const int lane = t & 31;
    const int wv   = t >> 5;

    const float* patch = tok2d + (size_t)b * (N_ + 1) * D_ + D_; // skip cls token

    float acc[9];
#pragma unroll
    for (int o = 0; o < 9; ++o) acc[o] = 0.f;

    for (int j = 0; j < 4; ++j) {
        const int d = t + j * 256;
        float conv = 0.f;
#pragma unroll
        for (int ky = 0; ky < 3; ++ky) {
            const int yy = y + ky - 1;
#pragma unroll
            for (int kx = 0; kx < 3; ++kx) {
                const int xx = x + kx - 1;
                float pv = 0.f;
                if (yy >= 0 && yy < H_ && xx >= 0 && xx < W_)
                    pv = patch[(size_t)(yy * W_ + xx) * D_ + d];
                conv = fmaf(pv, dw_w[d * 9 + ky * 3 + kx], conv);
            }
        }
#pragma unroll
        for (int o = 0; o < 9; ++o)
            acc[o] = fmaf(conv, pw_w[o * D_ + d], acc[o]);
    }

    __shared__ float red[8 * 9];
    __shared__ float logit[9];
#pragma unroll
    for (int o = 0; o < 9; ++o) {
        float v = acc[o];
        for (int s = 16; s > 0; s >>= 1) v += __shfl_down(v, s, 32);
        if (lane == 0) red[wv * 9 + o] = v;
    }
    __syncthreads();
    if (t < 9) {
        float s = 0.f;
        for (int w2 = 0; w2 < 8; ++w2) s += red[w2 * 9 + t];
        logit[t] = s + pw_b[t];
    }
    __syncthreads();
    if (t == 0) {
        float mx = logit[0];
        for (int o = 1; o < 9; ++o) mx = fmaxf(mx, logit[o]);
        float vals[9]; int cols[9];
#pragma unroll
        for (int o = 0; o < 9; ++o) {
            const int dy = o / 3 - 1, dx = o % 3 - 1;
            const int ny = min(max(y + dy, 0), H_ - 1);
            const int nx = min(max(x + dx, 0), W_ - 1);
            cols[o] = ny * W_ + nx;
            vals[o] = expf(logit[o] - mx);   // softmax norm cancels for argmax
        }
        // argmax over (duplicate-summed) neighbor columns; ties -> lowest col
        float best = -1.f; int bestc = N_;
        for (int o = 0; o < 9; ++o) {
            const int c = cols[o];
            bool first = true;
            for (int p = 0; p < o; ++p) if (cols[p] == c) { first = false; break; }
            if (!first) continue;
            float s = 0.f;
            for (int p = o; p < 9; ++p) if (cols[p] == c) s += vals[p];
            if (s > best || (s == best && c < bestc)) { best = s; bestc = c; }
        }
        parent[bn] = bestc;
    }
}

// ---------------------------------------------------------------------------
// Stage 2: 6 iterations of r = r[r] (pointer doubling) + deterministic counts.
// One block per batch, 576 threads.
// ---------------------------------------------------------------------------
__global__ __launch_bounds__(576)
void chase_counts_kernel(int* __restrict__ roots, float* __restrict__ counts)
{
    const int b = blockIdx.x;
    const int t = threadIdx.x;
    __shared__ int r0[N_];
    __shared__ int r1[N_];
    r0[t] = roots[b * N_ + t];
    __syncthreads();
    for (int it = 0; it < 6; ++it) {
        r1[t] = r0[r0[t]];
        __syncthreads();
        r0[t] = r1[t];
        __syncthreads();
    }
    roots[b * N_ + t] = r0[t];
    int c = 0;
    for (int nn = 0; nn < N_; ++nn) c += (r0[nn] == t) ? 1 : 0;
    counts[b * N_ + t] = (float)c;
}

// ---------------------------------------------------------------------------
// Stage 3: top-256 of counts (value desc, index asc ties, like lax.top_k),
// indices emitted in ascending order (matches sort(topk_idx)).
// ---------------------------------------------------------------------------
__global__ __launch_bounds__(576)
void topk_counts_kernel(const float* __restrict__ counts,
                        int* __restrict__ ord_idx,
                        float* __restrict__ cnt_sel)
{
    const int b = blockIdx.x;
    const int t = threadIdx.x;
    __shared__ float c[N_];
    __shared__ unsigned char fl[N_];
    c[t] = counts[b * N_ + t];
    __syncthreads();
    const float ci = c[t];
    int rank = 0;
    for (int j = 0; j < N_; ++j) {
        const float cj = c[j];
        rank += (cj > ci || (cj == ci && j < t)) ? 1 : 0;
    }
    const bool sel = rank < KSEL;
    fl[t] = sel ? 1 : 0;
    __syncthreads();
    if (sel) {
        int pos = 0;
        for (int j = 0; j < t; ++j) pos += fl[j];
        ord_idx[b * KSEL + pos] = t;
        cnt_sel[b * KSEL + pos] = ci;
    }
}

// ---------------------------------------------------------------------------
// Stage 4: cluster centers for the 256 selected roots, deterministic
// fixed-order accumulation (no atomics). One block per (b, slot).
// Thread t owns contiguous d in [4t, 4t+4) -> coalesced b128 traffic.
// ---------------------------------------------------------------------------
__global__ __launch_bounds__(256)
void centers_kernel(const float* __restrict__ tok2d,
                    const int* __restrict__ roots,
                    const int* __restrict__ ord_idx,
                    const float* __restrict__ cnt_sel,
                    float* __restrict__ C)
{
    const int bk = blockIdx.x;
    const int b  = bk / KSEL;
    const int k  = bk % KSEL;
    const int t  = threadIdx.x;
    __shared__ int r[N_];
    for (int nn = t; nn < N_; nn += 256) r[nn] = roots[b * N_ + nn];
    __syncthreads();
    const int   target = ord_idx[b * KSEL + k];
    const float cl     = fmaxf(cnt_sel[b * KSEL + k], 1.0f);
    const float* patch = tok2d + (size_t)b * (N_ + 1) * D_ + D_;
    float4 a = make_float4(0.f, 0.f, 0.f, 0.f);
    for (int nn = 0; nn < N_; ++nn) {
        if (r[nn] == target) {
            const float4 v = *(const float4*)(patch + (size_t)nn * D_ + 4 * t);
            a.x += v.x; a.y += v.y; a.z += v.z; a.w += v.w;
        }
    }
    float4 o;
    o.x = a.x / cl; o.y = a.y / cl; o.z = a.z / cl; o.w = a.w / cl;
    *(float4*)(C + ((size_t)b * KSEL + k) * D_ + 4 * t) = o;
}

// ---------------------------------------------------------------------------
// Stage 5: f32 WMMA GEMM, D = A(MxK) * B(KxN) (+bias), one 16x16 tile/wave.
// BT=1 -> B element (k,c) = Bsrc[c*ldb + k] (transposed access).
// V_WMMA_F32_16X16X4_F32 per ISA 7.12.2 layouts (wave32); BT is a template
// parameter so the inner loop is pure loads + wmma; two accumulator chains
// double the WMMA->WMMA RAW slack.
// ---------------------------------------------------------------------------
template <int BT>
__global__ __launch_bounds__(128)
void wmma_gemm_f32_kernel(const float* __restrict__ A, int lda,
                          const float* __restrict__ Bsrc, int ldb,
                          const float* __restrict__ bias,
                          float* __restrict__ Dm,
                          int M, int N, int K)
{
    const int lane = threadIdx.x & 31;
    const int wave = blockIdx.x * (blockDim.x >> 5) + (threadIdx.x >> 5);
    const int ntiles = N >> 4;
    const int mt = wave / ntiles;
    const int nt = wave % ntiles;
    if (mt * 16 >= M) return;              // wave-uniform guard (EXEC all-1s)
    const int hl = lane >> 4;              // lane half: K {0,1} vs {2,3}
    const int lr = lane & 15;
    const int arow = mt * 16 + lr;
    const int bcol = nt * 16 + lr;
    const float* Ap = A + (size_t)arow * lda + 2 * hl;   // +k per step
    v8f acc0 = {};
    v8f acc1 = {};
#pragma unroll 4
    for (int k = 0; k < K; k += 8) {
        // fragment pair for k .. k+3
        const int ka0 = k + 2 * hl;
        const v2f a0 = *(const v2f*)(Ap + k);
        v2f b0;
        if (BT) {
            b0 = *(const v2f*)(Bsrc + (size_t)bcol * ldb + ka0);
        } else {
            b0.x = Bsrc[(size_t)ka0 * ldb + bcol];
            b0.y = Bsrc[(size_t)(ka0 + 1) * ldb + bcol];
        }
        acc0 = __builtin_amdgcn_wmma_f32_16x16x4_f32(false, a0, false, b0,
                                                     (short)0, acc0, false, false);
        // fragment pair for k+4 .. k+7
        const int ka1 = k + 4 + 2 * hl;
        const v2f a1 = *(const v2f*)(Ap + k + 4);
        v2f b1;
        if (BT) {
            b1 = *(const v2f*)(Bsrc + (size_t)bcol * ldb + ka1);
        } else {
            b1.x = Bsrc[(size_t)ka1 * ldb + bcol];
            b1.y = Bsrc[(size_t)(ka1 + 1) * ldb + bcol];
        }
        acc1 = __builtin_amdgcn_wmma_f32_16x16x4_f32(false, a1, false, b1,
                                                     (short)0, acc1, false, false);
    }
    const float bb = bias ? bias[bcol] : 0.0f;
    const int rbase = mt * 16 + hl * 8;
#pragma unroll
    for (int i = 0; i < 8; ++i) {
        const int rrow = rbase + i;
        if (rrow < M) Dm[(size_t)rrow * N + bcol] = (acc0[i] + acc1[i]) + bb;
    }
}

// ---------------------------------------------------------------------------
// Stage 6: scores[b,k] = w[b] . C[b,k]  (constant offset/scale dropped: it
// does not change per-batch top-k ranking); masked -inf when count == 0.
// Thread t dots contiguous d in [4t, 4t+4) with b128 loads.
// ---------------------------------------------------------------------------
__global__ __launch_bounds__(256)
void scores_kernel(const float* __restrict__ wvec,
                   const float* __restrict__ C,
                   const float* __restrict__ cnt_sel,
                   float* __restrict__ scores)
{
    const int bk = blockIdx.x;
    const int b  = bk / KSEL;
    const int t  = threadIdx.x;
    const float4 wv = *(const float4*)(wvec + (size_t)b * D_ + 4 * t);
    const float4 cv = *(const float4*)(C + (size_t)bk * D_ + 4 * t);
    float s = wv.x * cv.x + wv.y * cv.y + wv.z * cv.z + wv.w * cv.w;
    __shared__ float red[256];
    red[t] = s;
    __syncthreads();
    for (int st = 128; st > 0; st >>= 1) {
        if (t < st) red[t] += red[t + st];
        __syncthreads();
    }
    if (t == 0) scores[bk] = (cnt_sel[bk] > 0.f) ? red[0] : -INFINITY;
}

// ---------------------------------------------------------------------------
// Stage 7: top-128 of 256 scores, ties (incl. -inf) -> lowest index,
// emitted ascending.
// ---------------------------------------------------------------------------
__global__ __launch_bounds__(256)
void select_kernel(const float* __restrict__ scores, int* __restrict__ sel_idx)
{
    const int b = blockIdx.x;
    const int t = threadIdx.x;
    __shared__ float sc[KSEL];
    __shared__ unsigned char fl[KSEL];
    sc[t] = scores[b * KSEL + t];
    __syncthreads();
    const float si = sc[t];
    int rank = 0;
    for (int j = 0; j < KSEL; ++j) {
        const float sj = sc[j];
        rank += (sj > si || (sj == si && j < t)) ? 1 : 0;
    }
    const bool sel = rank < MTOK;
    fl[t] = sel ? 1 : 0;
    __syncthreads();
    if (sel) {
        int pos = 0;
        for (int j = 0; j < t; ++j) pos += fl[j];
        sel_idx[b * MTOK + pos] = t;
    }
}

// ---------------------------------------------------------------------------
// Stage 8: gather selected C rows into the output (b128 copies).
// ---------------------------------------------------------------------------
__global__ __launch_bounds__(256)
void gather_kernel(const float* __restrict__ C, const int* __restrict__ sel_idx,
                   float* __restrict__ out)
{
    const size_t q = (size_t)blockIdx.x * 256 + threadIdx.x; // B*MTOK*D/4 exact
    const int dq = (int)(q % (D_ / 4));
    const int p  = (int)((q / (D_ / 4)) % MTOK);
    const int b  = (int)(q / ((size_t)(D_ / 4) * MTOK));
    const int k  = sel_idx[b * MTOK + p];
    const float4 v = *(const float4*)(C + ((size_t)b * KSEL + k) * D_ + 4 * dq);
    *(float4*)(out + q * 4) = v;
}

// ---------------------------------------------------------------------------
extern "C" void kernel_launch(void* const* d_in, const int* in_sizes, int n_in,
                              void* d_out, int out_size, void* d_ws, size_t ws_size,
                              hipStream_t stream)
{
    const float* tok2d = (const float*)d_in[0];
    const float* dw_w  = (const float*)d_in[1];
    const float* pw_w  = (const float*)d_in[2];
    const float* pw_b  = (const float*)d_in[3];
    const float* q_w   = (const float*)d_in[4];
    const float* q_b   = (const float*)d_in[5];
    const float* v_w   = (const float*)d_in[6];
    // d_in[7] = v_b: only shifts all scores of a batch equally -> ranking-invariant.
    // d_in[8] = K (256), d_in[9] = MAX_TOKENS (128): compile-time constants here.
    float* out = (float*)d_out;

    char* ws = (char*)d_ws;
    size_t off = 0;
    auto alloc = [&](size_t bytes) -> void* {
        void* p = ws + off;
        off += (bytes + 255) & ~(size_t)255;
        return p;
    };
    int*   roots   = (int*)  alloc((size_t)B_ * N_ * 4);
    float* counts  = (float*)alloc((size_t)B_ * N_ * 4);
    int*   ord_idx = (int*)  alloc((size_t)B_ * KSEL * 4);
    float* cnt_sel = (float*)alloc((size_t)B_ * KSEL * 4);
    float* Qcls    = (float*)alloc((size_t)B_ * D_ * 4);
    float* wvec    = (float*)alloc((size_t)B_ * D_ * 4);
    float* scoresb = (float*)alloc((size_t)B_ * KSEL * 4);
    int*   sel_idx = (int*)  alloc((size_t)B_ * MTOK * 4);
    float* Cbuf    = (float*)alloc((size_t)B_ * KSEL * D_ * 4);
    (void)ws_size; (void)in_sizes; (void)n_in; (void)out_size;

    affinity_parent_kernel<<<B_ * N_, 256, 0, stream>>>(tok2d, dw_w, pw_w, pw_b, roots);
    chase_counts_kernel<<<B_, N_, 0, stream>>>(roots, counts);
    topk_counts_kernel<<<B_, N_, 0, stream>>>(counts, ord_idx, cnt_sel);
    centers_kernel<<<B_ * KSEL, 256, 0, stream>>>(tok2d, roots, ord_idx, cnt_sel, Cbuf);

    // GEMM1: Qcls(32x1024) = cls(32x1024) @ q_w^T + q_b   (WMMA f32 16x16x4)
    {
        const int tiles = (B_ / 16) * (D_ / 16);   // 2 * 64 = 128 waves
        wmma_gemm_f32_kernel<1><<<tiles / 4, 128, 0, stream>>>(
            tok2d, (N_ + 1) * D_, q_w, D_, q_b, Qcls, B_, D_, D_);
    }
    // GEMM2: w(32x1024) = Qcls @ v_w   (replaces the 17-GFLOP K_v GEMM)
    {
        const int tiles = (B_ / 16) * (D_ / 16);
        wmma_gemm_f32_kernel<0><<<tiles / 4, 128, 0, stream>>>(
            Qcls, D_, v_w, D_, nullptr, wvec, B_, D_, D_);
    }

    scores_kernel<<<B_ * KSEL, 256, 0, stream>>>(wvec, Cbuf, cnt_sel, scoresb);
    select_kernel<<<B_, KSEL, 0, stream>>>(scoresb, sel_idx);
    gather_kernel<<<(B_ * MTOK * D_) / (256 * 4), 256, 0, stream>>>(Cbuf, sel_idx, out);
}